// SparseConvolutionDownsample_8581344657791
// MI455X (gfx1250) — compile-verified
//
#include <hip/hip_runtime.h>
#include <hip/hip_bf16.h>

typedef __attribute__((ext_vector_type(2))) float v2f;
typedef __attribute__((ext_vector_type(4))) float v4f;
typedef __attribute__((ext_vector_type(8))) float v8f;

#define CIN   128
#define COUT  256
#define NTAPS 4
#define BN_EPS 1e-4f

// ---- workspace layout (int granularity) ----
// ws[0..3]    : per-tap counts
// ws[4..7]    : per-tap scatter cursors
// ws[8..12]   : per-tap exclusive offsets (5 entries)
// ((float*)ws)[256..511]  : per-channel sum
// ((float*)ws)[512..767]  : per-channel sumsq
// ws[1024 ..] : perm[N] (row indices bucketed by tap)
#define WS_PERM_OFF 1024
#define WS_HEADER_BYTES 4096

// ---------------- bucketing ----------------
__global__ void sc_hist_k(const int* __restrict__ k_idx, int n, int* __restrict__ ws) {
    int i = blockIdx.x * blockDim.x + threadIdx.x;
    if (i < n) atomicAdd(&ws[k_idx[i]], 1);
}

__global__ void sc_scan_k(int* __restrict__ ws) {
    if (blockIdx.x == 0 && threadIdx.x == 0) {
        ws[8] = 0;
        #pragma unroll
        for (int k = 0; k < NTAPS; ++k) ws[8 + k + 1] = ws[8 + k] + ws[k];
    }
}

__global__ void sc_scatter_k(const int* __restrict__ k_idx, int n, int* __restrict__ ws) {
    int i = blockIdx.x * blockDim.x + threadIdx.x;
    if (i < n) {
        int k   = k_idx[i];
        int pos = ws[8 + k] + atomicAdd(&ws[4 + k], 1);
        ws[WS_PERM_OFF + pos] = i;
    }
}

// ---------------- WMMA gather-GEMM-scatter ----------------
// Block = 512 threads = 16 waves. One block handles one 16-row tile (all rows
// share tap k) times the full COUT=256 (wave w -> columns [16w, 16w+16)).
// K reduction = CIN = 128 -> 32 x V_WMMA_F32_16X16X4_F32 per tile.
__global__ __launch_bounds__(512)
void sc_gemm_wmma_k(const float* __restrict__ features,
                    const float* __restrict__ weight,   // [4][CIN][COUT]
                    const int*   __restrict__ seg_ids,
                    float*       __restrict__ out,      // [num_out][COUT], pre-zeroed
                    const int*   __restrict__ ws) {
    __shared__ float sA[16][CIN + 4];   // +4 pad: lane stride 132 -> conflict-free ds reads
    __shared__ int   sSeg[16];
    __shared__ int   sMeta[3];          // tap, rowbase, valid (tap = -1 -> no work)

    if (threadIdx.x == 0) {
        int t = blockIdx.x, rem = t, tap = -1;
        #pragma unroll
        for (int k = 0; k < NTAPS; ++k) {
            int tiles = (ws[k] + 15) >> 4;
            if (tap < 0) {
                if (rem < tiles) tap = k; else rem -= tiles;
            }
        }
        sMeta[0] = tap;
        if (tap >= 0) {
            sMeta[1] = ws[8 + tap] + rem * 16;
            int v = ws[tap] - rem * 16;
            sMeta[2] = v < 16 ? v : 16;
        }
    }
    __syncthreads();
    int tap = sMeta[0];
    if (tap < 0) return;                       // block-uniform exit
    int rowbase = sMeta[1];
    int valid   = sMeta[2];

    const int* perm = ws + WS_PERM_OFF;

    // Stage A tile: thread t loads row j = t/32, 4 floats at col (t%32)*4.
    {
        int j = threadIdx.x >> 5;
        int c = (threadIdx.x & 31) << 2;
        if (j < valid) {
            int row = perm[rowbase + j];
            v4f f = *(const v4f*)(features + (size_t)row * CIN + c);
            sA[j][c + 0] = f.x; sA[j][c + 1] = f.y;
            sA[j][c + 2] = f.z; sA[j][c + 3] = f.w;
            if (c == 0) sSeg[j] = seg_ids[row];
        } else {
            sA[j][c + 0] = 0.f; sA[j][c + 1] = 0.f;
            sA[j][c + 2] = 0.f; sA[j][c + 3] = 0.f;
            if (c == 0) sSeg[j] = -1;
        }
    }
    __syncthreads();

    int wave    = threadIdx.x >> 5;    // 0..15 -> column tile
    int lane    = threadIdx.x & 31;
    int half    = lane >> 4;           // 0: lanes 0-15, 1: lanes 16-31
    int lmod    = lane & 15;
    int colbase = wave << 4;

    // B fragment source: weight[tap][kb][colbase+lmod]; 512KB total -> L2-resident.
    const float* wb = weight + ((size_t)tap * CIN) * COUT + colbase + lmod;

    // ISA A layout (16x4 f32): lanes 0-15 M=0..15, VGPR0=K0/VGPR1=K1; lanes 16-31 K2/K3.
    // Mirrored B layout (4x16): lanes give N; VGPR+half give K.
    v8f acc = {};
    #pragma unroll
    for (int kk = 0; kk < CIN / 4; ++kk) {
        int kb = (kk << 2) + (half << 1);
        v2f a; a.x = sA[lmod][kb];            a.y = sA[lmod][kb + 1];
        v2f b; b.x = wb[(size_t)kb * COUT];   b.y = wb[(size_t)(kb + 1) * COUT];
        acc = __builtin_amdgcn_wmma_f32_16x16x4_f32(
            /*neg_a=*/false, a, /*neg_b=*/false, b,
            /*c_mod=*/(short)0, acc, /*reuse_a=*/false, /*reuse_b=*/false);
    }

    // C/D layout: lane holds N = lmod; VGPR v holds M = v + 8*half. Scatter-add.
    #pragma unroll
    for (int v = 0; v < 8; ++v) {
        int m = v + (half << 3);
        int s = sSeg[m];
        if (s >= 0)
            atomicAdd(out + (size_t)s * COUT + colbase + lmod, acc[v]);
    }
}

// ---------------- BatchNorm stats + normalize ----------------
__global__ void sc_stats_k(const float* __restrict__ out, int M, float* __restrict__ sums) {
    int c = threadIdx.x;                   // 256 channels
    float s = 0.f, s2 = 0.f;
    for (int r = blockIdx.x; r < M; r += gridDim.x) {
        float x = out[(size_t)r * COUT + c];
        s += x; s2 += x * x;
    }
    atomicAdd(&sums[c], s);
    atomicAdd(&sums[COUT + c], s2);
}

__global__ void sc_norm_k(float* __restrict__ out, int M,
                          const float* __restrict__ sums,
                          const float* __restrict__ gamma,
                          const float* __restrict__ beta) {
    __shared__ float s_scale[COUT], s_shift[COUT];
    int c = threadIdx.x;
    {
        float invM  = 1.f / (float)M;
        float mean  = sums[c] * invM;
        float var   = sums[COUT + c] * invM - mean * mean;
        float inv   = rsqrtf(var + BN_EPS);
        float scale = gamma[c] * inv;
        s_scale[c]  = scale;
        s_shift[c]  = beta[c] - mean * scale;
    }
    __syncthreads();
    size_t total = (size_t)M * COUT;
    for (size_t i = (size_t)blockIdx.x * blockDim.x + threadIdx.x;
         i < total; i += (size_t)gridDim.x * blockDim.x) {
        int ch  = (int)(i & (COUT - 1));
        float y = out[i] * s_scale[ch] + s_shift[ch];
        out[i]  = y > 0.f ? y : 0.f;
    }
}

// ---------------- launch ----------------
extern "C" void kernel_launch(void* const* d_in, const int* in_sizes, int n_in,
                              void* d_out, int out_size, void* d_ws, size_t ws_size,
                              hipStream_t stream) {
    const float* features = (const float*)d_in[0];
    const float* weight   = (const float*)d_in[1];
    const float* gamma    = (const float*)d_in[2];
    const float* beta     = (const float*)d_in[3];
    const int*   seg_ids  = (const int*)d_in[4];
    const int*   k_idx    = (const int*)d_in[5];

    int n = in_sizes[0] / CIN;     // N active input sites
    int M = out_size / COUT;       // num_out active output sites
    int*   ws  = (int*)d_ws;
    float* out = (float*)d_out;

    // zero accumulator + ws header (counts/cursors/offsets/sums) each call
    hipMemsetAsync(d_out, 0, (size_t)out_size * sizeof(float), stream);
    hipMemsetAsync(d_ws, 0, WS_HEADER_BYTES, stream);

    int b256 = (n + 255) / 256;
    sc_hist_k   <<<b256, 256, 0, stream>>>(k_idx, n, ws);
    sc_scan_k   <<<1, 32, 0, stream>>>(ws);
    sc_scatter_k<<<b256, 256, 0, stream>>>(k_idx, n, ws);

    int maxTiles = (n + 15) / 16 + NTAPS;   // upper bound incl. partial tiles per tap
    sc_gemm_wmma_k<<<maxTiles, 512, 0, stream>>>(features, weight, seg_ids, out, ws);

    float* sums = (float*)d_ws + 256;
    sc_stats_k<<<1024, COUT, 0, stream>>>(out, M, sums);
    sc_norm_k <<<2048, COUT, 0, stream>>>(out, M, sums, gamma, beta);
}